// StatefulChunk2_89326729822344
// MI455X (gfx1250) — compile-verified
//
#include <hip/hip_runtime.h>
#include <math.h>

typedef float v2f __attribute__((ext_vector_type(2)));
typedef float v8f __attribute__((ext_vector_type(8)));

#define NHEADS 8
#define HID    2048
#define FFD    8192
#define MAXHD  512
#define WCTX   512
#define FCTX   4096
#define PLIN   256
#define EPSV   1e-6f

// -------------------- gfx1250 async global->LDS (guarded) --------------------
#if defined(__has_builtin)
#if __has_builtin(__builtin_amdgcn_global_load_async_to_lds_b32) && \
    __has_builtin(__builtin_amdgcn_s_wait_asynccnt)
#define HAVE_ASYNC_LDS 1
#endif
#endif
#ifndef HAVE_ASYNC_LDS
#define HAVE_ASYNC_LDS 0
#endif

typedef __attribute__((address_space(1))) int g_i32;
typedef __attribute__((address_space(3))) int l_i32;

__device__ __forceinline__ void async_copy_f32(float* dst_lds, const float* src_glob) {
#if HAVE_ASYNC_LDS
  __builtin_amdgcn_global_load_async_to_lds_b32((g_i32*)src_glob, (l_i32*)dst_lds, 0, 0);
#else
  *dst_lds = *src_glob;
#endif
}
__device__ __forceinline__ void async_wait_all() {
#if HAVE_ASYNC_LDS
  __builtin_amdgcn_s_wait_asynccnt(0);
#endif
}

// -------------------- helpers --------------------

__device__ __forceinline__ float gelu_t(float x) {
  float x3 = x * x * x;
  return 0.5f * x * (1.0f + tanhf(0.7978845608028654f * (x + 0.044715f * x3)));
}

// Element (t,d) of the *updated* KV cache:
//  sliding: Kn = concat(Kc[1:], k_new)   -> t<W-1 ? Kc[t+1] : k_new
//  full:    Kn = Kc*(1-um) + k_new*um    (um is update_mask[t])
__device__ __forceinline__ float cache_elem(const float* __restrict__ C,
                                            const float* __restrict__ nw,
                                            const float* __restrict__ um,
                                            int is_full, int t, int d) {
  if (is_full) {
    float u = um[t];
    return C[(size_t)t * MAXHD + d] * (1.0f - u) + nw[d] * u;
  } else {
    return (t < WCTX - 1) ? C[(size_t)(t + 1) * MAXHD + d] : nw[d];
  }
}

// -------------------- elementwise / norm kernels --------------------

__global__ void copy_kernel(const float* __restrict__ in, float* __restrict__ out, int n) {
  int i = blockIdx.x * 256 + threadIdx.x;
  if (i < n) out[i] = in[i];
}

// out = in * rsqrt(mean(in^2)+eps) * (1+w)     (one block)
__global__ void rms_kernel(const float* __restrict__ in, const float* __restrict__ w,
                           float* __restrict__ out, int n) {
  __shared__ float red[256];
  int tid = threadIdx.x;
  float s = 0.f;
  for (int i = tid; i < n; i += 256) { float t = in[i]; s += t * t; }
  red[tid] = s; __syncthreads();
  for (int o = 128; o > 0; o >>= 1) { if (tid < o) red[tid] += red[tid + o]; __syncthreads(); }
  float r = rsqrtf(red[0] / (float)n + EPSV);
  for (int i = tid; i < n; i += 256) out[i] = in[i] * r * (1.0f + w[i]);
}

// h = (h + val * rsqrt(mean(val^2)+eps) * (1+w)) [* scale]   (one block)
__global__ void rms_add_kernel(float* __restrict__ h, const float* __restrict__ val,
                               const float* __restrict__ w, const float* __restrict__ scale,
                               int n) {
  __shared__ float red[256];
  int tid = threadIdx.x;
  float s = 0.f;
  for (int i = tid; i < n; i += 256) { float t = val[i]; s += t * t; }
  red[tid] = s; __syncthreads();
  for (int o = 128; o > 0; o >>= 1) { if (tid < o) red[tid] += red[tid + o]; __syncthreads(); }
  float r = rsqrtf(red[0] / (float)n + EPSV);
  float sc = scale ? scale[0] : 1.0f;
  for (int i = tid; i < n; i += 256) h[i] = (h[i] + val[i] * r * (1.0f + w[i])) * sc;
}

// g = gelu(g) * u
__global__ void gelu_mul_kernel(float* __restrict__ g, const float* __restrict__ u, int n) {
  int i = blockIdx.x * 256 + threadIdx.x;
  if (i < n) g[i] = gelu_t(g[i]) * u[i];
}

// t = gelu(t) * pl
__global__ void gelu_pl_kernel(float* __restrict__ t, const float* __restrict__ pl, int n) {
  int i = threadIdx.x;
  if (i < n) t[i] = gelu_t(t[i]) * pl[i];
}

// -------------------- GEMV: y[n] = sum_k x[k]*W[k*N+n] --------------------
// grid.x tiles N by 256; grid.y splits K (partial sums via global_atomic_add_f32;
// y must be zeroed when gridDim.y > 1). W reads are fully coalesced across lanes;
// activation slice staged to LDS via async-load-to-LDS when available; weight
// stream software-prefetched (global_prefetch_b8) since the row stride (N*4 B)
// defeats any hardware prefetcher.
__global__ void gemv_kernel(const float* __restrict__ x, const float* __restrict__ Wm,
                            float* __restrict__ y, int K, int N) {
  __shared__ float xs[2048];
  int kchunk = K / gridDim.y;
  int k0 = blockIdx.y * kchunk;
  for (int i = threadIdx.x; i < kchunk; i += 256) async_copy_f32(&xs[i], &x[k0 + i]);
  async_wait_all();
  __syncthreads();
  int n = blockIdx.x * 256 + threadIdx.x;
  if (n >= N) return;
  float acc = 0.f;
  const float* wp = Wm + (size_t)k0 * N + n;
  int kk = 0;
  for (; kk + 8 <= kchunk; kk += 8) {
    __builtin_prefetch(wp + (size_t)(kk + 24) * N, 0, 0);
#pragma unroll
    for (int j = 0; j < 8; ++j) acc = fmaf(xs[kk + j], wp[(size_t)(kk + j) * N], acc);
  }
  for (; kk < kchunk; ++kk) acc = fmaf(xs[kk], wp[(size_t)kk * N], acc);
  if (gridDim.y == 1) y[n] = acc;
  else atomicAdd(&y[n], acc);
}

// -------------------- q/k norm + RoPE, v norm --------------------
// blocks 0..7: q heads (rms+rope); block 8: k (rms+rope); block 9: v (v_norm)
__global__ void qkv_norm_rope_kernel(const float* __restrict__ q, const float* __restrict__ k,
                                     const float* __restrict__ v, const float* __restrict__ qn,
                                     const float* __restrict__ kn, const float* __restrict__ cosv,
                                     const float* __restrict__ sinv, float* __restrict__ qr,
                                     float* __restrict__ kr, float* __restrict__ vn, int hd) {
  __shared__ float red[256];
  int b = blockIdx.x, tid = threadIdx.x;
  const float* src; float* dst; const float* wgt; int do_rope = 1;
  if (b < NHEADS)      { src = q + (size_t)b * hd; dst = qr + (size_t)b * hd; wgt = qn; }
  else if (b == NHEADS){ src = k; dst = kr; wgt = kn; }
  else                 { src = v; dst = vn; wgt = nullptr; do_rope = 0; }
  float s = 0.f;
  for (int i = tid; i < hd; i += 256) { float t = src[i]; s += t * t; }
  red[tid] = s; __syncthreads();
  for (int o = 128; o > 0; o >>= 1) { if (tid < o) red[tid] += red[tid + o]; __syncthreads(); }
  float r = rsqrtf(red[0] / (float)hd + EPSV);
  int half = hd >> 1;
  for (int i = tid; i < hd; i += 256) {
    float xi = src[i] * r * (wgt ? (1.0f + wgt[i]) : 1.0f);
    if (!do_rope) { dst[i] = xi; continue; }
    int j = (i < half) ? i + half : i - half;
    float xj = src[j] * r * (wgt ? (1.0f + wgt[j]) : 1.0f);
    float rot = (i < half) ? -xj : xj;
    dst[i] = xi * cosv[i] + rot * sinv[i];
  }
}

// -------------------- WMMA attention: scores = q @ K^T --------------------
// One wave32 per 16-column tile of T. M=16 (8 heads + 8 zero rows), K=hd.
// K tile staged in LDS (computed blend); q rows staged via async-load-to-LDS.
__global__ void scores_wmma_kernel(const float* __restrict__ qr, const float* __restrict__ Kc,
                                   const float* __restrict__ knew, const float* __restrict__ um,
                                   int is_full, int hd, int hd_sh, int T,
                                   float* __restrict__ scores) {
  __shared__ float smem[12288];           // 16*hd (K tile) + 8*hd (q rows); 49KB @ hd=512
  float* ks = smem;
  float* qs = smem + 16 * hd;
  int t0 = blockIdx.x * 16;
  int lane = threadIdx.x;
  for (int idx = lane; idx < 8 * hd; idx += 32) async_copy_f32(&qs[idx], &qr[idx]);
  for (int idx = lane; idx < 16 * hd; idx += 32) {
    int r = idx >> hd_sh, c = idx & (hd - 1);
    ks[idx] = cache_elem(Kc, knew, um, is_full, t0 + r, c);
  }
  async_wait_all();
  __syncthreads();

  int half = lane >> 4, l15 = lane & 15;
  float am = (l15 < 8) ? 1.0f : 0.0f;     // zero-pad A rows 8..15 (no EXEC divergence)
  int arow = l15 & 7;
  v8f c = {0.f, 0.f, 0.f, 0.f, 0.f, 0.f, 0.f, 0.f};
  for (int kk = 0; kk < hd; kk += 4) {
    int kk0 = kk + half * 2;              // lanes 0-15: K=kk,kk+1 ; lanes 16-31: K=kk+2,kk+3
    v2f a, b;
    a.x = am * qs[arow * hd + kk0];
    a.y = am * qs[arow * hd + kk0 + 1];
    b.x = ks[l15 * hd + kk0];             // B[k][n] = K[t0+n][k]
    b.y = ks[l15 * hd + kk0 + 1];
    c = __builtin_amdgcn_wmma_f32_16x16x4_f32(false, a, false, b, (short)0, c, false, false);
  }
  if (half == 0) {                        // D: VGPR r, lanes 0-15 -> M=r (heads 0..7)
    for (int r = 0; r < 8; ++r) scores[(size_t)r * T + t0 + l15] = c[r];
  }
}

// -------------------- softmax over T per head (mask added) --------------------
__global__ void softmax_kernel(float* __restrict__ scores, const float* __restrict__ mask, int T) {
  __shared__ float red[256];
  int tid = threadIdx.x;
  float* row = scores + (size_t)blockIdx.x * T;
  float mx = -1e30f;
  for (int i = tid; i < T; i += 256) mx = fmaxf(mx, row[i] + mask[i]);
  red[tid] = mx; __syncthreads();
  for (int o = 128; o > 0; o >>= 1) { if (tid < o) red[tid] = fmaxf(red[tid], red[tid + o]); __syncthreads(); }
  mx = red[0]; __syncthreads();
  float s = 0.f;
  for (int i = tid; i < T; i += 256) { float e = __expf(row[i] + mask[i] - mx); row[i] = e; s += e; }
  red[tid] = s; __syncthreads();
  for (int o = 128; o > 0; o >>= 1) { if (tid < o) red[tid] += red[tid + o]; __syncthreads(); }
  float inv = 1.0f / red[0];
  for (int i = tid; i < T; i += 256) row[i] *= inv;
}

// -------------------- WMMA attention: out = attn @ V --------------------
// One wave32 per 16-column tile of hd. K=T tiled in 64-row LDS chunks.
// attn rows staged via async-load-to-LDS (per-lane addresses); V tile blended.
#define TCH 64
__global__ void attnv_wmma_kernel(const float* __restrict__ attn, const float* __restrict__ Vc,
                                  const float* __restrict__ vnew, const float* __restrict__ um,
                                  int is_full, int hd, int T, float* __restrict__ out) {
  __shared__ float vs[TCH * 16];
  __shared__ float as[8 * TCH];
  int d0 = blockIdx.x * 16;
  int lane = threadIdx.x;
  int half = lane >> 4, l15 = lane & 15;
  float am = (l15 < 8) ? 1.0f : 0.0f;
  int arow = l15 & 7;
  v8f c = {0.f, 0.f, 0.f, 0.f, 0.f, 0.f, 0.f, 0.f};
  for (int t = 0; t < T; t += TCH) {
    __syncthreads();
    for (int idx = lane; idx < 8 * TCH; idx += 32) {
      int r = idx >> 6, cc = idx & (TCH - 1);
      async_copy_f32(&as[idx], &attn[(size_t)r * T + t + cc]);
    }
    for (int idx = lane; idx < TCH * 16; idx += 32) {
      int r = idx >> 4, cc = idx & 15;
      vs[idx] = cache_elem(Vc, vnew, um, is_full, t + r, d0 + cc);
    }
    async_wait_all();
    __syncthreads();
    for (int kk = 0; kk < TCH; kk += 4) {
      int kk0 = kk + half * 2;
      v2f a, b;
      a.x = am * as[arow * TCH + kk0];
      a.y = am * as[arow * TCH + kk0 + 1];
      b.x = vs[kk0 * 16 + l15];           // B[k][n] = V[t+k][d0+n]
      b.y = vs[(kk0 + 1) * 16 + l15];
      c = __builtin_amdgcn_wmma_f32_16x16x4_f32(false, a, false, b, (short)0, c, false, false);
    }
  }
  if (half == 0) {
    for (int r = 0; r < 8; ++r) out[(size_t)r * hd + d0 + l15] = c[r];
  }
}

// -------------------- materialize updated KV slices for outputs --------------------
__global__ void kv_out_kernel(const float* __restrict__ C, const float* __restrict__ nw,
                              const float* __restrict__ um, int is_full, int hd,
                              float* __restrict__ out) {
  size_t idx = (size_t)blockIdx.x * 256 + threadIdx.x;
  int t = (int)(idx / hd), d = (int)(idx % hd);
  out[idx] = cache_elem(C, nw, um, is_full, t, d);
}

// -------------------- driver --------------------

extern "C" void kernel_launch(void* const* d_in, const int* in_sizes, int n_in,
                              void* d_out, int out_size, void* d_ws, size_t ws_size,
                              hipStream_t stream) {
  (void)in_sizes; (void)n_in; (void)out_size; (void)ws_size;
  const float* hidden     = (const float*)d_in[0];
  const float* mask_full  = (const float*)d_in[1];
  const float* mask_slide = (const float*)d_in[2];
  const float* um         = (const float*)d_in[3];
  const float* plc        = (const float*)d_in[4];
  const float* cos_s      = (const float*)d_in[5];
  const float* sin_s      = (const float*)d_in[6];
  const float* cos_f      = (const float*)d_in[7];
  const float* sin_f      = (const float*)d_in[8];
  const float* kvs        = (const float*)d_in[9];
  const float* kvf        = (const float*)d_in[10];

  float* out = (float*)d_out;
  float* ws  = (float*)d_ws;

  float* x  = ws + 0;        // 8192
  float* q  = ws + 8192;     // 4096
  float* k  = ws + 12288;    // 512
  float* v  = ws + 12800;    // 512
  float* qr = ws + 13312;    // 4096
  float* kr = ws + 17408;    // 512
  float* vn = ws + 17920;    // 512
  float* sc = ws + 18432;    // 8*4096 (scores)
  float* ao = ws + 83968;    // 4096 (attn out)
  float* pj = ws + 88064;    // 2048
  float* g  = ws + 90112;    // 8192
  float* u  = ws + 98304;    // 8192
  float* m  = ws + 106496;   // 2048
  float* t1 = ws + 108544;   // 256
  float* t2 = ws + 108800;   // 2048

  float* h      = out;                               // final hidden lives in d_out
  float* out13k = out + 2048;
  float* out13v = out13k + (size_t)WCTX * 256;
  float* out14k = out13v + (size_t)WCTX * 256;
  float* out14v = out14k + (size_t)FCTX * 512;

  copy_kernel<<<HID / 256, 256, 0, stream>>>(hidden, h, HID);

  const int lidx[7]  = {8, 9, 10, 11, 12, 13, 14};
  const int lfull[7] = {0, 1, 0, 0, 0, 0, 1};
  int si = 0, fi = 0;

  for (int L = 0; L < 7; ++L) {
#define PP(j) ((const float*)d_in[11 + L * 17 + (j)])
    const float *Wq = PP(0), *Wk = PP(1), *Wv = PP(2), *Wo = PP(3), *qn = PP(4), *kn = PP(5),
                *ln_in = PP(6), *ln_pa = PP(7), *ln_pre = PP(8), *ln_post = PP(9),
                *ln_pl = PP(10), *Wg = PP(11), *Wu = PP(12), *Wd = PP(13), *Wpg = PP(14),
                *Wpp = PP(15), *scale = PP(16);
#undef PP
    int full = lfull[L];
    int hd = full ? 512 : 256;
    int hd_sh = full ? 9 : 8;
    int T  = full ? FCTX : WCTX;
    const float* cosv = full ? cos_f : cos_s;
    const float* sinv = full ? sin_f : sin_s;
    const float* Kc = full ? kvf + (size_t)fi * FCTX * MAXHD : kvs + (size_t)si * WCTX * MAXHD;
    const float* Vc = full ? kvf + (size_t)(2 + fi) * FCTX * MAXHD
                           : kvs + (size_t)(5 + si) * WCTX * MAXHD;
    const float* mask = full ? mask_full : mask_slide;
    int NQ = NHEADS * hd;

    // ---- attention block ----
    rms_kernel<<<1, 256, 0, stream>>>(h, ln_in, x, HID);
    (void)hipMemsetAsync(q, 0, (size_t)NQ * sizeof(float), stream);
    gemv_kernel<<<dim3(NQ / 256, 8), 256, 0, stream>>>(x, Wq, q, HID, NQ);
    (void)hipMemsetAsync(k, 0, (size_t)hd * sizeof(float), stream);
    gemv_kernel<<<dim3(hd / 256, 8), 256, 0, stream>>>(x, Wk, k, HID, hd);
    (void)hipMemsetAsync(v, 0, (size_t)hd * sizeof(float), stream);
    gemv_kernel<<<dim3(hd / 256, 8), 256, 0, stream>>>(x, Wv, v, HID, hd);
    qkv_norm_rope_kernel<<<NHEADS + 2, 256, 0, stream>>>(q, k, v, qn, kn, cosv, sinv,
                                                         qr, kr, vn, hd);

    if (lidx[L] == 13) {  // sliding layer: export Kn/Vn[..., :256]
      kv_out_kernel<<<(WCTX * 256) / 256, 256, 0, stream>>>(Kc, kr, um, 0, 256, out13k);
      kv_out_kernel<<<(WCTX * 256) / 256, 256, 0, stream>>>(Vc, vn, um, 0, 256, out13v);
    }
    if (lidx[L] == 14) {  // full layer: export Kn/Vn[..., :512]
      kv_out_kernel<<<(FCTX * 512) / 256, 256, 0, stream>>>(Kc, kr, um, 1, 512, out14k);
      kv_out_kernel<<<(FCTX * 512) / 256, 256, 0, stream>>>(Vc, vn, um, 1, 512, out14v);
    }

    scores_wmma_kernel<<<T / 16, 32, 0, stream>>>(qr, Kc, kr, um, full, hd, hd_sh, T, sc);
    softmax_kernel<<<NHEADS, 256, 0, stream>>>(sc, mask, T);
    attnv_wmma_kernel<<<hd / 16, 32, 0, stream>>>(sc, Vc, vn, um, full, hd, T, ao);

    (void)hipMemsetAsync(pj, 0, (size_t)HID * sizeof(float), stream);
    gemv_kernel<<<dim3(HID / 256, 8), 256, 0, stream>>>(ao, Wo, pj, NQ, HID);
    rms_add_kernel<<<1, 256, 0, stream>>>(h, pj, ln_pa, nullptr, HID);

    // ---- MLP block ----
    rms_kernel<<<1, 256, 0, stream>>>(h, ln_pre, x, HID);
    (void)hipMemsetAsync(g, 0, (size_t)FFD * sizeof(float), stream);
    gemv_kernel<<<dim3(FFD / 256, 8), 256, 0, stream>>>(x, Wg, g, HID, FFD);
    (void)hipMemsetAsync(u, 0, (size_t)FFD * sizeof(float), stream);
    gemv_kernel<<<dim3(FFD / 256, 8), 256, 0, stream>>>(x, Wu, u, HID, FFD);
    gelu_mul_kernel<<<FFD / 256, 256, 0, stream>>>(g, u, FFD);
    (void)hipMemsetAsync(m, 0, (size_t)HID * sizeof(float), stream);
    gemv_kernel<<<dim3(HID / 256, 8), 256, 0, stream>>>(g, Wd, m, FFD, HID);
    rms_add_kernel<<<1, 256, 0, stream>>>(h, m, ln_post, nullptr, HID);

    // ---- per-layer block (+ scale) ----
    (void)hipMemsetAsync(t1, 0, (size_t)PLIN * sizeof(float), stream);
    gemv_kernel<<<dim3(1, 8), 256, 0, stream>>>(h, Wpg, t1, HID, PLIN);
    gelu_pl_kernel<<<1, 256, 0, stream>>>(t1, plc + lidx[L] * PLIN, PLIN);
    gemv_kernel<<<dim3(HID / 256, 1), 256, 0, stream>>>(t1, Wpp, t2, PLIN, HID);
    rms_add_kernel<<<1, 256, 0, stream>>>(h, t2, ln_pl, scale, HID);

    if (full) fi++; else si++;
  }
}